// ReadoutLayer_1855425872033
// MI455X (gfx1250) — compile-verified
//
#include <hip/hip_runtime.h>

typedef __attribute__((ext_vector_type(2))) float v2f;
typedef __attribute__((ext_vector_type(8))) float v8f;

#define BATCH   128
#define NMAX    512
#define MMAX    512
#define DIM     128
#define NEG_INF (-1e9f)
#define SCALE   0.08838834764831845f   // 1/sqrt(128)

// One workgroup = 8 wave32 waves, handles one (batch b, 16-row M tile).
// Each wave computes 4 N-tiles of 16 cols (64 cols), full K=128 via
// v_wmma_f32_16x16x4_f32, then a fused masked log-softmax over N=512.
__launch_bounds__(256)
__global__ void readout_logsoftmax_kernel(const float* __restrict__ res,
                                          const float* __restrict__ hsqc,
                                          const int*   __restrict__ n_res,
                                          const int*   __restrict__ n_hsqc,
                                          float*       __restrict__ out) {
    const int b    = blockIdx.x / (MMAX / 16);
    const int mt   = blockIdx.x % (MMAX / 16);
    const int m0   = mt * 16;
    const int wave = threadIdx.x >> 5;
    const int lane = threadIdx.x & 31;
    const int hl   = lane >> 4;      // half-wave select (0/1)
    const int l16  = lane & 15;

    const int nres  = n_res[b];
    const int nhsqc = n_hsqc[b];

    const float* hb = hsqc + (size_t)b * MMAX * DIM;   // A source: 16 rows of M
    const float* rb = res  + (size_t)b * NMAX * DIM;   // B source: res rows (B = res^T)

    // ---- GEMM: 4 accumulators of 16x16 f32 per wave --------------------
    v8f C[4];
    #pragma unroll
    for (int j = 0; j < 4; ++j)
        #pragma unroll
        for (int e = 0; e < 8; ++e) C[j][e] = 0.0f;

    // A fragment: lane holds hsqc[m0+l16][k + 2*hl .. +1]  (ISA 16x4 f32 A layout)
    const float* aptr = hb + (size_t)(m0 + l16) * DIM + 2 * hl;
    const int nbase = wave * 64;

    #pragma unroll 4
    for (int k = 0; k < DIM; k += 4) {
        v2f a = *(const v2f*)(aptr + k);
        #pragma unroll
        for (int j = 0; j < 4; ++j) {
            const int n0 = nbase + j * 16;
            // B fragment: B[kk][n] = res[n][kk]; lane holds res[n0+l16][k + 2*hl .. +1]
            v2f bf = *(const v2f*)(rb + (size_t)(n0 + l16) * DIM + 2 * hl + k);
            C[j] = __builtin_amdgcn_wmma_f32_16x16x4_f32(
                false, a, false, bf, (short)0, C[j], false, false);
        }
    }

    // ---- scale + residue-column mask (in registers) --------------------
    // C layout: VGPR r, lanes 0-15 -> (M=r, N=l16), lanes 16-31 -> (M=r+8, N=l16)
    float s[4][8];
    #pragma unroll
    for (int j = 0; j < 4; ++j) {
        const int n = nbase + j * 16 + l16;
        #pragma unroll
        for (int r = 0; r < 8; ++r)
            s[j][r] = (n < nres) ? C[j][r] * SCALE : NEG_INF;
    }

    __shared__ float wred[8][16];   // per-wave row partials
    __shared__ float grow[16];      // reduced per-row value (max, then lse)

    // ---- pass 1: global row max ---------------------------------------
    float gmax[8];
    #pragma unroll
    for (int r = 0; r < 8; ++r) {
        float m = s[0][r];
        #pragma unroll
        for (int j = 1; j < 4; ++j) m = fmaxf(m, s[j][r]);
        #pragma unroll
        for (int off = 1; off < 16; off <<= 1)          // stays inside 16-lane row group
            m = fmaxf(m, __shfl_xor(m, off, 32));
        if (l16 == 0) wred[wave][r + 8 * hl] = m;
    }
    __syncthreads();
    if (threadIdx.x < 16) {
        float m = wred[0][threadIdx.x];
        #pragma unroll
        for (int w = 1; w < 8; ++w) m = fmaxf(m, wred[w][threadIdx.x]);
        grow[threadIdx.x] = m;
    }
    __syncthreads();
    #pragma unroll
    for (int r = 0; r < 8; ++r) gmax[r] = grow[r + 8 * hl];

    // ---- pass 2: global row log-sum-exp -------------------------------
    #pragma unroll
    for (int r = 0; r < 8; ++r) {
        float acc = 0.0f;
        #pragma unroll
        for (int j = 0; j < 4; ++j) acc += __expf(s[j][r] - gmax[r]);
        #pragma unroll
        for (int off = 1; off < 16; off <<= 1)
            acc += __shfl_xor(acc, off, 32);
        if (l16 == 0) wred[wave][r + 8 * hl] = acc;
    }
    __syncthreads();
    if (threadIdx.x < 16) {
        float acc = wred[0][threadIdx.x];
        #pragma unroll
        for (int w = 1; w < 8; ++w) acc += wred[w][threadIdx.x];
        grow[threadIdx.x] = __logf(acc);
    }
    __syncthreads();
    float lse[8];
    #pragma unroll
    for (int r = 0; r < 8; ++r) lse[r] = grow[r + 8 * hl];

    // ---- write logp with validity mask (streaming stores) -------------
    const size_t obase = ((size_t)b * MMAX + m0) * NMAX;
    #pragma unroll
    for (int j = 0; j < 4; ++j) {
        const int n = nbase + j * 16 + l16;
        #pragma unroll
        for (int r = 0; r < 8; ++r) {
            const int m = r + 8 * hl;
            float v = 0.0f;
            if ((m0 + m) < nhsqc && n < nres)
                v = s[j][r] - gmax[r] - lse[r];
            __builtin_nontemporal_store(v, out + obase + (size_t)m * NMAX + n);
        }
    }
}

extern "C" void kernel_launch(void* const* d_in, const int* in_sizes, int n_in,
                              void* d_out, int out_size, void* d_ws, size_t ws_size,
                              hipStream_t stream) {
    const float* res   = (const float*)d_in[0];   // res_emb  [B, NMAX, D]
    const float* hsqc  = (const float*)d_in[1];   // hsqc_emb [B, MMAX, D]
    const int*   nres  = (const int*)d_in[2];     // n_res  [B]
    const int*   nhsqc = (const int*)d_in[3];     // n_hsqc [B]
    float*       out   = (float*)d_out;           // [B, MMAX, NMAX]

    dim3 grid(BATCH * (MMAX / 16));
    dim3 block(256);
    readout_logsoftmax_kernel<<<grid, block, 0, stream>>>(res, hsqc, nres, nhsqc, out);
}